// EmbeddingAttrs_5428838662424
// MI455X (gfx1250) — compile-verified
//
#include <hip/hip_runtime.h>

// Embedding gather + concat:
//   out[row] = concat(W_elem[chem[row]] (64f), W_res[res[row]] (32f),
//                     charge[row] (16f), W_kano[node[row]] (300f))  -> 412 floats/row
//
// Pure streaming-store workload: 824 MB out @ 23.3 TB/s HBM => ~37us floor.
// Strategy: one float4 segment per thread; output index == global work index
// => single contiguous 128-bit store stream with NT temporal hint.

typedef float v4f __attribute__((ext_vector_type(4)));

#define SEGS_PER_ROW 103u   // 412 floats / 4
// float4-section layout within a row:
//   [0,16)  elem   (W_elem row = 16 float4s)
//   [16,24) res    (W_res  row =  8 float4s)
//   [24,28) charge (        row =  4 float4s)
//   [28,103) kano  (W_kano row = 75 float4s)

__global__ __launch_bounds__(256) void embed_concat_kernel(
    const int* __restrict__ chem,
    const int* __restrict__ resi,
    const v4f* __restrict__ charge,   // [N*4] float4
    const int* __restrict__ node,
    const v4f* __restrict__ Welem,    // [256*16] float4
    const v4f* __restrict__ Wres,     // [32*8] float4
    const v4f* __restrict__ Wkano,    // [119*75] float4
    v4f* __restrict__ out,            // [N*103] float4
    long total)                       // N * 103
{
    long stride = (long)gridDim.x * blockDim.x;
    for (long i = (long)blockIdx.x * blockDim.x + threadIdx.x; i < total; i += stride) {
        unsigned ii  = (unsigned)i;                 // < 51.5M, fits u32
        unsigned row = ii / SEGS_PER_ROW;           // mulhi + shift
        unsigned seg = ii - row * SEGS_PER_ROW;

        v4f v;
        if (seg < 16u) {
            // element embedding: tiny table, keep cached (RT load)
            v = Welem[(unsigned)chem[row] * 16u + seg];
        } else if (seg < 24u) {
            v = Wres[(unsigned)resi[row] * 8u + (seg - 16u)];
        } else if (seg < 28u) {
            // charge is streamed read-once: non-temporal load
            v = __builtin_nontemporal_load(&charge[row * 4u + (seg - 24u)]);
        } else {
            v = Wkano[(unsigned)node[row] * 75u + (seg - 28u)];
        }

        // Output is touch-once 824MB stream: NT store (global_store_b128 th:NT),
        // address == i*16B => perfectly contiguous across the grid.
        __builtin_nontemporal_store(v, &out[i]);
    }
}

extern "C" void kernel_launch(void* const* d_in, const int* in_sizes, int n_in,
                              void* d_out, int out_size, void* d_ws, size_t ws_size,
                              hipStream_t stream) {
    // setup_inputs() order:
    //   0 chem_element (int32 [N])
    //   1 residue_type (int32 [N])
    //   2 charge       (f32   [N,16])
    //   3 node_types   (int32 [N])
    //   4 W_elem       (f32   [256,64])
    //   5 W_res        (f32   [32,32])
    //   6 W_kano       (f32   [119,300])
    const int* chem   = (const int*)d_in[0];
    const int* resi   = (const int*)d_in[1];
    const v4f* charge = (const v4f*)d_in[2];
    const int* node   = (const int*)d_in[3];
    const v4f* Welem  = (const v4f*)d_in[4];
    const v4f* Wres   = (const v4f*)d_in[5];
    const v4f* Wkano  = (const v4f*)d_in[6];
    v4f* out = (v4f*)d_out;

    long n = in_sizes[0];                 // N_ATOMS
    long total = n * (long)SEGS_PER_ROW;  // float4 segments

    const int threads = 256;              // 8 wave32s per block
    long blocks = (total + threads - 1) / threads;
    if (blocks > 0x7FFFFFFFL) blocks = 0x7FFFFFFFL;  // grid-stride covers rest

    embed_concat_kernel<<<(unsigned)blocks, threads, 0, stream>>>(
        chem, resi, charge, node, Welem, Wres, Wkano, out, total);
}